// GraphGatherMol_89489938579864
// MI455X (gfx1250) — compile-verified
//
#include <hip/hip_runtime.h>
#include <hip/hip_bf16.h>

// Problem constants (match reference)
#define BATCH     4096
#define MAX_ATOMS 128
#define N_FEAT    128

typedef __attribute__((ext_vector_type(2))) float v2f;
typedef __attribute__((ext_vector_type(8))) float v8f;

// One molecule per wave32. Masked atom-sum done as WMMA f32 16x16x4:
//   D(16x16) += A(16x4 mask, rows identical) * B(4x16 = 4 atom rows x 16 feats)
// 8 accumulator chains (feature chunks of 16) x 32 K-steps (atom chunks of 4).
// All rows of D are identical, so acc[c][0] at lane j holds feature 16c+j.
__global__ __launch_bounds__(256) void GraphGatherMol_wmma_kernel(
    const float* __restrict__ x,   // [BATCH, MAX_ATOMS, N_FEAT]
    const int*   __restrict__ ds,  // [BATCH, 2] : (valid_atoms, valid_feats)
    float*       __restrict__ out) // [BATCH, N_FEAT]
{
    const int lane = threadIdx.x & 31;
    const int wave = threadIdx.x >> 5;
    const int mol  = blockIdx.x * 8 + wave;
    if (mol >= BATCH) return;          // wave-uniform

    const int va = ds[mol * 2 + 0];    // valid atoms (1..128), uniform in wave
    const int vf = ds[mol * 2 + 1];    // valid feats (1..128), uniform in wave

    const int h = lane >> 4;           // 0: lanes 0-15 (K=0,1) ; 1: lanes 16-31 (K=2,3)
    const int j = lane & 15;           // N / feature-within-chunk index

    const float* __restrict__ base = x + (size_t)mol * (MAX_ATOMS * N_FEAT);
    // per-lane pointer at (row = 2h, col = j) of this molecule's tile
    const float* __restrict__ p = base + h * 2 * N_FEAT + j;

    v8f acc[8] = {};

    // Uniform early exit: rows >= va are fully masked -> skip their traffic.
    for (int a0 = 0; a0 < va; a0 += 4) {
        // A-matrix: atom mask, per documented 16x4 f32 A layout
        v2f am;
        am.x = (a0 + 2 * h     < va) ? 1.0f : 0.0f;
        am.y = (a0 + 2 * h + 1 < va) ? 1.0f : 0.0f;

        const float* __restrict__ q = p + (size_t)a0 * N_FEAT;

#pragma unroll
        for (int c = 0; c < 8; ++c) {
            // B-matrix: lanes 0-15 rows K=0,1 ; lanes 16-31 rows K=2,3
            // (coalesced: 16 consecutive lanes read 16 consecutive floats)
            v2f bm;
            bm.x = q[c * 16];           // X[a0 + 2h,     16c + j]
            bm.y = q[c * 16 + N_FEAT];  // X[a0 + 2h + 1, 16c + j]

            acc[c] = __builtin_amdgcn_wmma_f32_16x16x4_f32(
                /*neg_a=*/false, am,
                /*neg_b=*/false, bm,
                /*c_mod=*/(short)0, acc[c],
                /*reuse_a=*/false, /*reuse_b=*/false);
        }
    }

    // Epilogue: every element of D equals the reduction result for its column.
    // Lane j (j<16) owns features {16c + j}. Apply feature mask + ReLU.
    if (lane < 16) {
        float* __restrict__ o = out + (size_t)mol * N_FEAT;
#pragma unroll
        for (int c = 0; c < 8; ++c) {
            const int f = c * 16 + j;
            float v = acc[c][0];
            v = (f < vf) ? fmaxf(v, 0.0f) : 0.0f;
            o[f] = v;
        }
    }
}

extern "C" void kernel_launch(void* const* d_in, const int* in_sizes, int n_in,
                              void* d_out, int out_size, void* d_ws, size_t ws_size,
                              hipStream_t stream) {
    (void)in_sizes; (void)n_in; (void)d_ws; (void)ws_size; (void)out_size;

    const float* node_features = (const float*)d_in[0]; // [4096,128,128] f32
    const int*   data_slice    = (const int*)d_in[1];   // [4096,2] i32
    float*       out           = (float*)d_out;         // [4096,128] f32

    // 512 blocks x 256 threads = 4096 wave32 = one wave per molecule
    dim3 grid(BATCH / 8);
    dim3 block(256);
    GraphGatherMol_wmma_kernel<<<grid, block, 0, stream>>>(node_features, data_slice, out);
}